// Int4MatmulDML_52544629899327
// MI455X (gfx1250) — compile-verified
//
#include <hip/hip_runtime.h>
#include <hip/hip_bf16.h>

typedef __attribute__((ext_vector_type(2))) float v2f;
typedef __attribute__((ext_vector_type(8))) float v8f;

#define TOKENS_    4096
#define INF_       4096
#define OUTF_      11008
#define BM         128
#define BN         64
#define BK         32
#define XS_STRIDE  36    // 128 rows * 36 floats: 16B-aligned rows, conflict-free A reads
#define WSP_STRIDE 160   // 16 kpair-rows * 160 floats: pair-interleaved B tile,
                         // stride%64==32 -> half-waves hit disjoint bank halves

__global__ __launch_bounds__(256)
void int4_gemm_wmma_f32(const float* __restrict__ x,
                        const unsigned char* __restrict__ wp,
                        const float* __restrict__ scale,
                        const float* __restrict__ zero,
                        float* __restrict__ out)
{
    __shared__ float Xs[BM * XS_STRIDE];          // 18432 B
    __shared__ float Ws[(BK / 2) * WSP_STRIDE];   // 10240 B, [kpair][2n + (k&1)]

    const int tid   = threadIdx.x;
    const int lane  = tid & 31;
    const int wave  = tid >> 5;
    const int l16   = lane & 15;
    const int lhalf = lane >> 4;           // 0: lanes 0-15, 1: lanes 16-31

    const int m_blk = blockIdx.y * BM;
    const int n_blk = blockIdx.x * BN;

    // ---- W loader mapping: 4 threads per output row, each handles 4 packed bytes (8 k's)
    const int wn = tid >> 2;               // local n, 0..63
    const int wg = tid & 3;                // which 4-byte group of the 16-byte K-chunk
    const float s_n = scale[n_blk + wn];
    const float z_n = zero[n_blk + wn];
    const unsigned char* wrow = wp + (size_t)(n_blk + wn) * (INF_ / 2);

    // ---- X loader mapping: 8 threads per row (float4 each), 32 rows per pass, 4 passes
    const int xr = tid >> 3;               // 0..31
    const int xc = (tid & 7) * 4;          // float column within K-step

    // ---- wave tile: 8 waves as 4(M) x 2(N), each wave owns 32x32
    const int wm  = (wave >> 1) * 32;
    const int wn0 = (wave & 1) * 32;

    v8f acc[2][2];
#pragma unroll
    for (int i = 0; i < 2; ++i)
#pragma unroll
        for (int j = 0; j < 2; ++j)
            acc[i][j] = (v8f){0.f, 0.f, 0.f, 0.f, 0.f, 0.f, 0.f, 0.f};

    for (int kb = 0; kb < INF_; kb += BK) {
        // ---- stage x tile (128 x 32 f32), fully coalesced 128B/row
#pragma unroll
        for (int p = 0; p < 4; ++p) {
            const int row = xr + p * 32;
            const float4 v = *(const float4*)(x + (size_t)(m_blk + row) * INF_ + kb + xc);
            float* dst = &Xs[row * XS_STRIDE + xc];
            dst[0] = v.x; dst[1] = v.y; dst[2] = v.z; dst[3] = v.w;
        }
        // ---- stage + dequant W tile, pair-interleaved: Ws[kpair][2n] = k-even, [2n+1] = k-odd
        {
            const unsigned int packed = *(const unsigned int*)(wrow + (kb >> 1) + wg * 4);
#pragma unroll
            for (int i = 0; i < 4; ++i) {
                const unsigned int b = (packed >> (8 * i)) & 0xFFu;
                const int kp = wg * 4 + i;                 // local kpair, 0..15
                v2f wv;
                wv.x = ((float)(b & 15u) - z_n) * s_n;     // k = 2*kp
                wv.y = ((float)(b >> 4)  - z_n) * s_n;     // k = 2*kp + 1
                *(v2f*)&Ws[kp * WSP_STRIDE + 2 * wn] = wv; // single ds_store_b64
            }
        }
        // ---- prefetch next K-step's global lines (global_prefetch_b8)
        if (kb + BK < INF_) {
            __builtin_prefetch(x + (size_t)(m_blk + xr) * INF_ + (kb + BK) + xc, 0, 0);
            __builtin_prefetch(wrow + ((kb + BK) >> 1) + wg * 4, 0, 0);
        }
        __syncthreads();

        // ---- 8 k-substeps x 2x2 tiles = 32 WMMAs per K-step per wave
#pragma unroll
        for (int k = 0; k < BK; k += 4) {
            v2f a[2], b[2];
#pragma unroll
            for (int mi = 0; mi < 2; ++mi) {
                // A 16x4: lane L holds {A[M=L%16][k + 2*(L/16)], A[..][k+1+2*(L/16)]}
                const float* ap = &Xs[(wm + mi * 16 + l16) * XS_STRIDE + k + 2 * lhalf];
                a[mi].x = ap[0];
                a[mi].y = ap[1];
            }
#pragma unroll
            for (int ni = 0; ni < 2; ++ni) {
                // B 4x16: lane needs {B[k+2*lhalf][n], B[k+1+2*lhalf][n]} — adjacent
                // in the pair-interleaved layout -> one aligned ds_load_b64
                const int kp = (k >> 1) + lhalf;
                b[ni] = *(const v2f*)&Ws[kp * WSP_STRIDE + 2 * (wn0 + ni * 16 + l16)];
            }
#pragma unroll
            for (int mi = 0; mi < 2; ++mi)
#pragma unroll
                for (int ni = 0; ni < 2; ++ni)
                    acc[mi][ni] = __builtin_amdgcn_wmma_f32_16x16x4_f32(
                        false, a[mi], false, b[ni],
                        (short)0, acc[mi][ni], false, false);
        }
        __syncthreads();
    }

    // ---- write out: C VGPR r holds M = r + 8*(lane/16), N = lane%16
#pragma unroll
    for (int mi = 0; mi < 2; ++mi) {
#pragma unroll
        for (int ni = 0; ni < 2; ++ni) {
            const int Nc = n_blk + wn0 + ni * 16 + l16;
#pragma unroll
            for (int r = 0; r < 8; ++r) {
                const int Mr = m_blk + wm + mi * 16 + 8 * lhalf + r;
                out[(size_t)Mr * OUTF_ + Nc] = acc[mi][ni][r];
            }
        }
    }
}

extern "C" void kernel_launch(void* const* d_in, const int* in_sizes, int n_in,
                              void* d_out, int out_size, void* d_ws, size_t ws_size,
                              hipStream_t stream) {
    const float*         x    = (const float*)d_in[0];
    const unsigned char* wpk  = (const unsigned char*)d_in[1];
    const float*         scl  = (const float*)d_in[2];
    const float*         zro  = (const float*)d_in[3];
    float*               out  = (float*)d_out;

    dim3 grid(OUTF_ / BN, TOKENS_ / BM);   // 172 x 32
    int4_gemm_wmma_f32<<<grid, 256, 0, stream>>>(x, wpk, scl, zro, out);
}